// GlobalAttentionPool_21964462752171
// MI455X (gfx1250) — compile-verified
//
#include <hip/hip_runtime.h>
#include <hip/hip_bf16.h>
#include <cstdint>

#define HIDDEN 128
#define NGRAPH 256

typedef float v2f __attribute__((ext_vector_type(2)));
typedef float v8f __attribute__((ext_vector_type(8)));

// ---------------- K0: zero-init workspace + output ----------------
__global__ void k0_init(float* sagg, unsigned* gmaxU, float* gsum,
                        int* nstart, int* nend, float* out, int N) {
    int i = blockIdx.x * blockDim.x + threadIdx.x;
    if (i < N) sagg[i] = 0.0f;
    if (i < NGRAPH) {
        gmaxU[i]  = 0u;          // below the order-key of every float
        gsum[i]   = 0.0f;
        nstart[i] = 0x7FFFFFFF;
        nend[i]   = -1;
    }
    if (i < NGRAPH * HIDDEN) out[i] = 0.0f;
}

// ---------------- K1: p = x@w_rel, r = x@w_root via V_WMMA_F32_16X16X4_F32 ----
// One wave32 per 16-row tile of x. A: 16x4 f32 (documented layout:
// M=lane&15; VGPR0/1 hold K = 4k+2*half + {0,1}). B: 4x16 with column 0 =
// w_rel slice, column 1 = w_root slice (N = lane&15, K striped like A).
// D (16x16 f32, documented): c[j] lanes 0-15 -> D[j][lane], lanes 16-31 ->
// D[8+j][lane-16]; we keep columns N=0 (p) and N=1 (r).
__global__ void __launch_bounds__(256)
k1_gemv_wmma(const float* __restrict__ x,
             const float* __restrict__ w_rel,
             const float* __restrict__ w_root,
             float* __restrict__ p, float* __restrict__ r,
             int N, int numTiles) {
    __shared__ float swr[HIDDEN];
    __shared__ float swo[HIDDEN];
    int tid = threadIdx.x;
    if (tid < HIDDEN) swr[tid] = w_rel[tid];
    else              swo[tid - HIDDEN] = w_root[tid - HIDDEN];
    __syncthreads();

    int wave = tid >> 5;
    int lane = tid & 31;
    int tile = blockIdx.x * 8 + wave;
    if (tile >= numTiles) return;          // wave-uniform: EXEC all-1 for WMMA
    int base = tile * 16;

    int m    = lane & 15;                  // A: row in tile / B: column N
    int half = lane >> 4;
    int row  = base + m;
    bool rowOK = row < N;
    const float* xr = x + (size_t)row * HIDDEN;

    v8f c = {0.f, 0.f, 0.f, 0.f, 0.f, 0.f, 0.f, 0.f};
#pragma unroll
    for (int k = 0; k < HIDDEN / 4; ++k) {
        int kk = 4 * k + 2 * half;
        v2f a;
        a.x = rowOK ? xr[kk]     : 0.0f;
        a.y = rowOK ? xr[kk + 1] : 0.0f;
        v2f b;
        b.x = 0.0f; b.y = 0.0f;
        if (m == 0)      { b.x = swr[kk]; b.y = swr[kk + 1]; }   // col 0: w_rel
        else if (m == 1) { b.x = swo[kk]; b.y = swo[kk + 1]; }   // col 1: w_root
        c = __builtin_amdgcn_wmma_f32_16x16x4_f32(false, a, false, b,
                                                  (short)0, c, false, false);
    }

    if (m < 2) {
        float* dst = (m == 0) ? p : r;
        int rbase = base + half * 8;
#pragma unroll
        for (int j = 0; j < 8; ++j) {
            int rr = rbase + j;
            if (rr < N) dst[rr] = c[j];
        }
    }
}

// ---------------- K2: sagg[dst] += p[src] over all edges ----------------
__global__ void k2_edges(const int* __restrict__ src, const int* __restrict__ dst,
                         const float* __restrict__ p, float* __restrict__ sagg,
                         int E) {
    int e = blockIdx.x * blockDim.x + threadIdx.x;
    if (e < E) atomicAdd(&sagg[dst[e]], p[src[e]]);
}

// order-preserving float<->uint keys for atomic max
__device__ __forceinline__ unsigned fkey(float f) {
    unsigned b = __float_as_uint(f);
    return (b & 0x80000000u) ? ~b : (b | 0x80000000u);
}
__device__ __forceinline__ float funkey(unsigned u) {
    return (u & 0x80000000u) ? __uint_as_float(u ^ 0x80000000u)
                             : __uint_as_float(~u);
}

// ---------------- K3: logits, segment max, graph boundaries ----------------
__global__ void k3_logits(const float* __restrict__ sagg,
                          const float* __restrict__ r,
                          const float* __restrict__ b_rel,
                          const int* __restrict__ batch,
                          float* __restrict__ s, unsigned* __restrict__ gmaxU,
                          int* __restrict__ nstart, int* __restrict__ nend,
                          int N) {
    int n = blockIdx.x * blockDim.x + threadIdx.x;
    if (n >= N) return;
    int g = batch[n];
    float sv = sagg[n] + r[n] + b_rel[0];
    s[n] = sv;
    atomicMax(&gmaxU[g], fkey(sv));
    if (n == 0     || batch[n - 1] != g) nstart[g] = n;   // batch is sorted
    if (n == N - 1 || batch[n + 1] != g) nend[g]   = n;
}

// ---------------- K4: e = exp(s - max), segment sum ----------------
__global__ void k4_exp(const int* __restrict__ batch,
                       const unsigned* __restrict__ gmaxU,
                       float* __restrict__ s, float* __restrict__ gsum, int N) {
    int n = blockIdx.x * blockDim.x + threadIdx.x;
    if (n >= N) return;
    int g = batch[n];
    float e = expf(s[n] - funkey(gmaxU[g]));
    s[n] = e;                                   // overwrite logits with e
    atomicAdd(&gsum[g], e);
}

// ---------------- K5: gx[g] = sum_n (e[n]/denom[g]) * x[n] ----------------
// One block per graph; batch sorted => nodes [nstart..nend] all belong to g.
__global__ void __launch_bounds__(256)
k5_pool(const float* __restrict__ x, const float* __restrict__ s,
        const float* __restrict__ gsum, const int* __restrict__ nstart,
        const int* __restrict__ nend, float* __restrict__ out) {
    int g = blockIdx.x;
    int tid = threadIdx.x;
    int st = nstart[g], en = nend[g];
    __shared__ float red[256];
    float acc = 0.0f;
    if (st <= en) {
        float inv = 1.0f / gsum[g];
        int col  = tid & (HIDDEN - 1);
        int part = tid >> 7;
        for (int n = st + part; n <= en; n += 2)
            acc += x[(size_t)n * HIDDEN + col] * (s[n] * inv);
    }
    red[tid] = acc;
    __syncthreads();
    if (tid < HIDDEN)
        out[(size_t)g * HIDDEN + tid] = red[tid] + red[tid + HIDDEN];
}

extern "C" void kernel_launch(void* const* d_in, const int* in_sizes, int n_in,
                              void* d_out, int out_size, void* d_ws, size_t ws_size,
                              hipStream_t stream) {
    const float* x      = (const float*)d_in[0];
    const float* w_rel  = (const float*)d_in[1];
    const float* b_rel  = (const float*)d_in[2];
    const float* w_root = (const float*)d_in[3];
    const int*   eidx   = (const int*)d_in[4];
    const int*   batch  = (const int*)d_in[5];
    float* out = (float*)d_out;

    int N = in_sizes[0] / HIDDEN;
    int E = in_sizes[4] / 2;

    char* w = (char*)d_ws;
    float*    p      = (float*)w;    w += (size_t)N * 4;
    float*    r      = (float*)w;    w += (size_t)N * 4;
    float*    sagg   = (float*)w;    w += (size_t)N * 4;
    float*    s      = (float*)w;    w += (size_t)N * 4;
    unsigned* gmaxU  = (unsigned*)w; w += (size_t)NGRAPH * 4;
    float*    gsum   = (float*)w;    w += (size_t)NGRAPH * 4;
    int*      nstart = (int*)w;      w += (size_t)NGRAPH * 4;
    int*      nend   = (int*)w;      w += (size_t)NGRAPH * 4;

    k0_init<<<(N + 255) / 256, 256, 0, stream>>>(sagg, gmaxU, gsum, nstart, nend, out, N);

    int numTiles = (N + 15) / 16;
    k1_gemv_wmma<<<(numTiles + 7) / 8, 256, 0, stream>>>(x, w_rel, w_root, p, r, N, numTiles);

    k2_edges<<<(E + 255) / 256, 256, 0, stream>>>(eidx, eidx + E, p, sagg, E);

    k3_logits<<<(N + 255) / 256, 256, 0, stream>>>(sagg, r, b_rel, batch, s,
                                                   gmaxU, nstart, nend, N);

    k4_exp<<<(N + 255) / 256, 256, 0, stream>>>(batch, gmaxU, s, gsum, N);

    k5_pool<<<NGRAPH, 256, 0, stream>>>(x, s, gsum, nstart, nend, out);
}